// HMPNN_2568390443510
// MI455X (gfx1250) — compile-verified
//
#include <hip/hip_runtime.h>

typedef __attribute__((ext_vector_type(2))) float v2f;
typedef __attribute__((ext_vector_type(8))) float v8f;

#define GEMM_BLOCK 128   // 4 waves (wave32)

__device__ __forceinline__ float sigmoidf(float x) {
    return 1.0f / (1.0f + __expf(-x));
}

__device__ __forceinline__ v8f wmma_f32(v2f a, v2f b, v8f c) {
    return __builtin_amdgcn_wmma_f32_16x16x4_f32(
        /*neg_a=*/false, a, /*neg_b=*/false, b,
        /*c_mod=*/(short)0, c, /*reuse_a=*/false, /*reuse_b=*/false);
}

// D[M x N] = act(A[M x (K1+K2)] @ B[(K1+K2) x N] + bias)
// A row = [A1 row (K1 floats) || A2 row (K2 floats)]  (A2 unused if K2==0)
//
// Compile-time K1,K2,N -> fully unrolled K-loop, immediate DS / global
// offsets off hoisted base pointers (no per-iteration address math).
// Each wave: 2x2 tile block = 32 rows x 32 cols, 4 v8f accumulators;
// per k-step: 2 ds_load_b64 (A) + 4 global_load_b32 (B) -> 4 WMMAs.
// Block (4 waves) covers 32 rows x 128 cols. Rows clamped on load,
// masked on store, so M need not divide 32. EXEC stays all-ones.
template <int K1, int K2, int N, int ACT>
__global__ __launch_bounds__(GEMM_BLOCK) void hmpnn_gemm_wmma(
    const float* __restrict__ A1, const float* __restrict__ A2,
    const float* __restrict__ B, const float* __restrict__ bias,
    float* __restrict__ D, int M)
{
    constexpr int K  = K1 + K2;
    constexpr int LS = K + 4;   // LDS row stride: % 64 == 4 -> conflict-free
    __shared__ float sA[32 * LS];

    const int tid   = threadIdx.x;
    const int wave  = tid >> 5;
    const int lane  = tid & 31;
    const int tileM = blockIdx.x * 32;

    // Warm the (block-shared) weight panel: global_prefetch_b8.
    for (int k = tid; k < K; k += GEMM_BLOCK)
        __builtin_prefetch(B + (size_t)k * N, 0, 0);

    // Cooperative float4 staging of the 32 x K A-tile (rows clamped).
    constexpr int n4row = K >> 2;
    for (int idx = tid; idx < 32 * n4row; idx += GEMM_BLOCK) {
        int row = idx / n4row;
        int c   = (idx - row * n4row) << 2;
        int r   = tileM + row;
        if (r > M - 1) r = M - 1;
        float4 v;
        if (c < K1) v = *(const float4*)(A1 + (size_t)r * K1 + c);
        else        v = *(const float4*)(A2 + (size_t)r * K2 + (c - K1));
        *(float4*)(&sA[row * LS + c]) = v;
    }
    __syncthreads();

    const int half = lane >> 4;          // 0: lanes 0-15, 1: lanes 16-31
    const int rowA = lane & 15;
    const int c0   = wave * 32 + (lane & 15);   // this lane's column, tile 0
    const int c1   = c0 + 16;                   // tile 1
    const int cl0  = (c0 < N) ? c0 : (N - 1);   // clamped for loads
    const int cl1  = (c1 < N) ? c1 : (N - 1);

    // Hoisted bases: the k-loop below uses only immediate offsets.
    const float* sA0 = &sA[rowA * LS + half * 2];          // rows  0-15
    const float* sA1 = &sA[(rowA + 16) * LS + half * 2];   // rows 16-31
    const float* B0  = B + (size_t)(half * 2) * N + cl0;
    const float* B1  = B + (size_t)(half * 2) * N + cl1;

    v8f acc00 = {}, acc01 = {}, acc10 = {}, acc11 = {};
    #pragma unroll
    for (int k = 0; k < K; k += 4) {
        v2f a0 = *(const v2f*)(sA0 + k);
        v2f a1 = *(const v2f*)(sA1 + k);
        v2f b0, b1;
        b0.x = B0[(size_t)k * N];
        b0.y = B0[(size_t)(k + 1) * N];
        b1.x = B1[(size_t)k * N];
        b1.y = B1[(size_t)(k + 1) * N];
        acc00 = wmma_f32(a0, b0, acc00);
        acc01 = wmma_f32(a0, b1, acc01);
        acc10 = wmma_f32(a1, b0, acc10);
        acc11 = wmma_f32(a1, b1, acc11);
    }

    const float bv0 = bias[cl0];
    const float bv1 = bias[cl1];
    #pragma unroll
    for (int r = 0; r < 8; ++r) {
        const int row0 = tileM + r + half * 8;   // row tile 0
        const int row1 = row0 + 16;              // row tile 1
        float v00 = acc00[r] + bv0, v01 = acc01[r] + bv1;
        float v10 = acc10[r] + bv0, v11 = acc11[r] + bv1;
        if (ACT) { v00 = sigmoidf(v00); v01 = sigmoidf(v01);
                   v10 = sigmoidf(v10); v11 = sigmoidf(v11); }
        if (row0 < M) {
            if (c0 < N) D[(size_t)row0 * N + c0] = v00;
            if (c1 < N) D[(size_t)row0 * N + c1] = v01;
        }
        if (row1 < M) {
            if (c0 < N) D[(size_t)row1 * N + c0] = v10;
            if (c1 < N) D[(size_t)row1 * N + c1] = v11;
        }
    }
}

__global__ void hmpnn_zero_f4(float4* __restrict__ p, int n4)
{
    int i = blockIdx.x * blockDim.x + threadIdx.x;
    if (i < n4) {
        float4 z = {0.f, 0.f, 0.f, 0.f};
        p[i] = z;
    }
}

// dst[sidx[i]] += vals[i] * src[gidx[i]]  over 128-float rows.
// One wave per incidence, 4 floats per lane, global_atomic_add_f32.
__global__ __launch_bounds__(256) void hmpnn_scatter_add(
    const float* __restrict__ src, const int* __restrict__ gidx,
    const int* __restrict__ sidx, const float* __restrict__ vals,
    float* __restrict__ dst, int n)
{
    int inc = blockIdx.x * 8 + (threadIdx.x >> 5);
    if (inc >= n) return;
    int lane = threadIdx.x & 31;
    int g = gidx[inc];
    int s = sidx[inc];
    float v = vals[inc];
    const float4 x = *(const float4*)(src + (size_t)g * 128 + lane * 4);
    float* dp = dst + (size_t)s * 128 + lane * 4;
    atomicAdd(dp + 0, v * x.x);
    atomicAdd(dp + 1, v * x.y);
    atomicAdd(dp + 2, v * x.z);
    atomicAdd(dp + 3, v * x.w);
}

// x = sigmoid(x + m), vectorized float4.
__global__ void hmpnn_residual_sigmoid4(float4* __restrict__ x,
                                        const float4* __restrict__ m, int n4)
{
    int i = blockIdx.x * blockDim.x + threadIdx.x;
    if (i < n4) {
        float4 a = x[i];
        float4 b = m[i];
        a.x = sigmoidf(a.x + b.x);
        a.y = sigmoidf(a.y + b.y);
        a.z = sigmoidf(a.z + b.z);
        a.w = sigmoidf(a.w + b.w);
        x[i] = a;
    }
}

extern "C" void kernel_launch(void* const* d_in, const int* in_sizes, int n_in,
                              void* d_out, int out_size, void* d_ws, size_t ws_size,
                              hipStream_t stream)
{
    const float* x_0      = (const float*)d_in[0];
    const float* x_1      = (const float*)d_in[1];
    const int*   node_idx = (const int*)d_in[2];
    const int*   he_idx   = (const int*)d_in[3];
    const float* inc_vals = (const float*)d_in[4];
    const float* W_enc    = (const float*)d_in[5];
    const float* b_enc    = (const float*)d_in[6];
    const float* W_msg0   = (const float*)d_in[7];
    const float* b_msg0   = (const float*)d_in[8];
    const float* W_msg1   = (const float*)d_in[9];
    const float* b_msg1   = (const float*)d_in[10];
    const float* W_out    = (const float*)d_in[11];
    const float* b_out    = (const float*)d_in[12];
    float* out = (float*)d_out;

    const int N_NODES = 100000, N_HE = 50000, N_INC = 600000;
    const int DH = 128;

    // Workspace layout (all f32): ~205 MB total.
    float* ws      = (float*)d_ws;
    float* h0      = ws;                                // N_NODES * DH
    float* h1      = h0 + (size_t)N_NODES * DH;         // N_HE * DH
    float* mbuf    = h1 + (size_t)N_HE * DH;            // N_NODES * DH (m0, then m1)
    float* m0_he   = mbuf + (size_t)N_NODES * DH;       // N_HE * DH
    float* m1_node = m0_he + (size_t)N_HE * DH;         // N_NODES * DH

    const dim3 gblk(GEMM_BLOCK);
    const int gridN  = (N_NODES + 31) / 32;   // 3125
    const int gridHe = (N_HE + 31) / 32;      // 1563 (last block row-clamped)
    const int nN4  = N_NODES * DH / 4;
    const int nHe4 = N_HE * DH / 4;

    // Encode: h = x @ W_enc + b_enc
    hmpnn_gemm_wmma<256, 0, 128, 0><<<gridN, gblk, 0, stream>>>(
        x_0, nullptr, W_enc, b_enc, h0, N_NODES);
    hmpnn_gemm_wmma<256, 0, 128, 0><<<gridHe, gblk, 0, stream>>>(
        x_1, nullptr, W_enc, b_enc, h1, N_HE);

    for (int l = 0; l < 2; ++l) {
        const float* Wm0 = W_msg0 + (size_t)l * DH * DH;
        const float* bm0 = b_msg0 + (size_t)l * DH;
        const float* Wm1 = W_msg1 + (size_t)l * 2 * DH * DH;
        const float* bm1 = b_msg1 + (size_t)l * DH;

        // m0 = sigmoid(h0 @ Wm0 + bm0)
        hmpnn_gemm_wmma<128, 0, 128, 1><<<gridN, gblk, 0, stream>>>(
            h0, nullptr, Wm0, bm0, mbuf, N_NODES);

        // m0_he = segment_sum(inc_vals * m0[node_idx], he_idx)
        hmpnn_zero_f4<<<(nHe4 + 255) / 256, 256, 0, stream>>>((float4*)m0_he, nHe4);
        hmpnn_scatter_add<<<(N_INC + 7) / 8, 256, 0, stream>>>(
            mbuf, node_idx, he_idx, inc_vals, m0_he, N_INC);

        // m1 = sigmoid([h1 || m0_he] @ Wm1 + bm1)
        hmpnn_gemm_wmma<128, 128, 128, 1><<<gridHe, gblk, 0, stream>>>(
            h1, m0_he, Wm1, bm1, mbuf, N_HE);

        // m1_node = segment_sum(inc_vals * m1[he_idx], node_idx)
        hmpnn_zero_f4<<<(nN4 + 255) / 256, 256, 0, stream>>>((float4*)m1_node, nN4);
        hmpnn_scatter_add<<<(N_INC + 7) / 8, 256, 0, stream>>>(
            mbuf, he_idx, node_idx, inc_vals, m1_node, N_INC);

        // Updates
        hmpnn_residual_sigmoid4<<<(nN4 + 255) / 256, 256, 0, stream>>>(
            (float4*)h0, (const float4*)m1_node, nN4);
        hmpnn_residual_sigmoid4<<<(nHe4 + 255) / 256, 256, 0, stream>>>(
            (float4*)h1, (const float4*)m0_he, nHe4);
    }

    // out = h0 @ W_out + b_out  (N=40: stores masked)
    hmpnn_gemm_wmma<128, 0, 40, 0><<<gridN, gblk, 0, stream>>>(
        h0, nullptr, W_out, b_out, out, N_NODES);
}